// AuxiliaryModel_57707180589353
// MI455X (gfx1250) — compile-verified
//
#include <hip/hip_runtime.h>

typedef float v2f __attribute__((ext_vector_type(2)));
typedef float v8f __attribute__((ext_vector_type(8)));

#define BATCH   256
#define LSEQ    1024
#define USEQ    256
#define CH      16
#define STRIDE  1034      // 8*STRIDE % 64 == 16 -> bank-conflict-free dual-half access
#define NPOOL   5760      // pooled length per channel
#define XROW    1280      // L + U
#define EPSV    1e-5f

__global__ __launch_bounds__(256)
void tree_conv_pool_kernel(const float* __restrict__ x,
                           const float* __restrict__ leaf_w,  const float* __restrict__ leaf_b,
                           const float* __restrict__ unm_w,   const float* __restrict__ unm_b,
                           const float* __restrict__ conv_w,  const float* __restrict__ conv_b,
                           const float* __restrict__ bn_gamma,const float* __restrict__ bn_beta,
                           const float* __restrict__ bn_mean, const float* __restrict__ bn_var,
                           const float* __restrict__ leaf_nw, const float* __restrict__ internal_nw,
                           float* __restrict__ out)
{
    __shared__ float buf[2 * CH * STRIDE];   // ping-pong level buffers: [c][t] with row stride
    __shared__ float s_nw[1024];             // internal node weights

    const int b    = blockIdx.x;
    const int tid  = threadIdx.x;
    const int lane = tid & 31;               // wave32
    const int wave = tid >> 5;               // 8 waves
    const int nn   = lane & 15;              // column (time within tile) / A-row m
    const int hi   = lane >> 4;              // half-wave selector

    // stage internal_nw into LDS (broadcast reads later)
    for (int p = tid; p < 1023; p += 256) s_nw[p] = internal_nw[p];

    // --- per-lane constant A fragments: af[k][q], A layout 16x4 f32 ---
    // lane: M = nn ; VGPR0 = K=2*hi, VGPR1 = K=2*hi+1 ; global K = q*4 + local K
    v2f af[3][4];
    #pragma unroll
    for (int k = 0; k < 3; ++k) {
        #pragma unroll
        for (int q = 0; q < 4; ++q) {
            const int ci0 = q * 4 + hi * 2;
            af[k][q].x = conv_w[(nn * 16 + ci0    ) * 3 + k];
            af[k][q].y = conv_w[(nn * 16 + ci0 + 1) * 3 + k];
        }
    }
    // --- per-lane BN epilogue constants for rows m = r + 8*hi ---
    float asc8[8], bsh8[8];
    #pragma unroll
    for (int r = 0; r < 8; ++r) {
        const int m = r + 8 * hi;
        const float s = bn_gamma[m] * rsqrtf(bn_var[m] + EPSV);
        asc8[r] = s;
        bsh8[r] = (conv_b[m] - bn_mean[m]) * s + bn_beta[m];
    }

    float* const buf0 = buf;
    float* const buf1 = buf + CH * STRIDE;
    float* const outB = out + (size_t)b * (CH * NPOOL);

    // ---- Leaf stage: buf0[c][l] = (x*lw + lb) * leaf_nw ----
    for (int l = tid; l < LSEQ; l += 256) {
        const float xv = x[b * XROW + l];
        const float nw = leaf_nw[l];
        #pragma unroll
        for (int c = 0; c < CH; ++c)
            buf0[c * STRIDE + l] = (xv * leaf_w[l * 16 + c] + leaf_b[l * 16 + c]) * nw;
    }
    __syncthreads();

    // ---- pooled leaf segment (j offset 0) ----
    for (int p = tid; p < CH * 512; p += 256) {
        const int c = p >> 9, j = p & 511;
        outB[c * NPOOL + j] = fmaxf(buf0[c * STRIDE + 2 * j], buf0[c * STRIDE + 2 * j + 1]);
    }
    // ---- unmatched segment (j offset 512), no LDS needed ----
    for (int p = tid; p < CH * 128; p += 256) {
        const int c = p >> 7, j = p & 127;
        const int u0 = 2 * j;
        const float v0 = x[b * XROW + LSEQ + u0    ] * unm_w[ u0      * 16 + c] + unm_b[ u0      * 16 + c];
        const float v1 = x[b * XROW + LSEQ + u0 + 1] * unm_w[(u0 + 1) * 16 + c] + unm_b[(u0 + 1) * 16 + c];
        outB[c * NPOOL + 512 + j] = fmaxf(v0, v1);
    }
    // no barrier needed: level 0 only reads buf0 / writes buf1

    // ---- 10 tree levels ----
    int par = 0, nwoff = 0;
    for (int lev = 0; lev < 10; ++lev) {
        float* const curp = par ? buf1 : buf0;
        float* const nxtp = par ? buf0 : buf1;
        const int segmask = (2 << lev) - 1;    // segment length 2^(lev+1)

        // 64 tiles of 16 time-steps; wave-uniform distribution (EXEC stays full)
        for (int tile = wave; tile < 64; tile += 8) {
            const int t = tile * 16 + nn;
            const bool leftOK  = (t & segmask) != 0;
            const bool rightOK = (t & segmask) != segmask;
            // clamped (always in-range) neighbor columns; zeros selected afterwards
            const int tl = leftOK  ? t - 1 : t;
            const int tr = rightOK ? t + 1 : t;

            // ---- gather all 12 B fragments first (unconditional ds_loads + cndmask) ----
            v2f bf[3][4];
            #pragma unroll
            for (int q = 0; q < 4; ++q) {
                const int ci0 = q * 4 + hi * 2;
                const int r0  =  ci0      * STRIDE;
                const int r1  = (ci0 + 1) * STRIDE;
                const float l0 = curp[r0 + tl], l1 = curp[r1 + tl];
                const float m0 = curp[r0 + t ], m1 = curp[r1 + t ];
                const float r0v = curp[r0 + tr], r1v = curp[r1 + tr];
                bf[0][q].x = leftOK  ? l0  : 0.f;  bf[0][q].y = leftOK  ? l1  : 0.f;
                bf[1][q].x = m0;                   bf[1][q].y = m1;
                bf[2][q].x = rightOK ? r0v : 0.f;  bf[2][q].y = rightOK ? r1v : 0.f;
            }

            // ---- 12 chained WMMAs: D += W(k,q) x X(k,q) ----
            v8f acc = {0.f, 0.f, 0.f, 0.f, 0.f, 0.f, 0.f, 0.f};
            #pragma unroll
            for (int k = 0; k < 3; ++k) {
                #pragma unroll
                for (int q = 0; q < 4; ++q) {
                    acc = __builtin_amdgcn_wmma_f32_16x16x4_f32(false, af[k][q], false, bf[k][q],
                                                                (short)0, acc, false, false);
                }
            }

            // epilogue: BN(+conv_b) fused, ReLU, node scale; store rows m = r + 8*hi
            const float nwv = s_nw[nwoff + (t >> (lev + 1))];
            #pragma unroll
            for (int r = 0; r < 8; ++r) {
                const int m = r + 8 * hi;
                nxtp[m * STRIDE + t] = fmaxf(acc[r] * asc8[r] + bsh8[r], 0.f) * nwv;
            }
        }
        __syncthreads();

        // pooled output for this level (j offset 640 + lev*512)
        const int segoff = 640 + lev * 512;
        for (int p = tid; p < CH * 512; p += 256) {
            const int c = p >> 9, j = p & 511;
            outB[c * NPOOL + segoff + j] =
                fmaxf(nxtp[c * STRIDE + 2 * j], nxtp[c * STRIDE + 2 * j + 1]);
        }
        __syncthreads();

        nwoff += (512 >> lev);
        par ^= 1;
    }
}

extern "C" void kernel_launch(void* const* d_in, const int* in_sizes, int n_in,
                              void* d_out, int out_size, void* d_ws, size_t ws_size,
                              hipStream_t stream) {
    (void)in_sizes; (void)n_in; (void)d_ws; (void)ws_size; (void)out_size;
    const float* x           = (const float*)d_in[0];
    const float* leaf_w      = (const float*)d_in[1];
    const float* leaf_b      = (const float*)d_in[2];
    const float* unm_w       = (const float*)d_in[3];
    const float* unm_b       = (const float*)d_in[4];
    const float* conv_w      = (const float*)d_in[5];
    const float* conv_b      = (const float*)d_in[6];
    const float* bn_gamma    = (const float*)d_in[7];
    const float* bn_beta     = (const float*)d_in[8];
    const float* bn_mean     = (const float*)d_in[9];
    const float* bn_var      = (const float*)d_in[10];
    const float* leaf_nw     = (const float*)d_in[11];
    const float* internal_nw = (const float*)d_in[12];
    float* out = (float*)d_out;

    tree_conv_pool_kernel<<<dim3(BATCH), dim3(256), 0, stream>>>(
        x, leaf_w, leaf_b, unm_w, unm_b, conv_w, conv_b,
        bn_gamma, bn_beta, bn_mean, bn_var, leaf_nw, internal_nw, out);
}